// RigidGroupProjector_87239375716829
// MI455X (gfx1250) — compile-verified
//
#include <hip/hip_runtime.h>
#include <hip/hip_bf16.h>

typedef __attribute__((ext_vector_type(2))) float v2f;
typedef __attribute__((ext_vector_type(8))) float v8f;

#define GS    64            // group size (fixed by problem)
#define GPW   5             // (b,g) pairs handled per wave (5*3 = 15 rows of the 16x16 tile)
#define WAVES 4             // waves per block
#define TPB   (WAVES * 32)
#define NONES (GS * 3)      // ones region: covers dword offsets 3*n0 + {0..5}, n0 < 64

// ---------------- 3x3 Kabsch rotation from cross-covariance H ----------------
// H[i][j] = sum_n Yc[n][i] * Xc[n][j].  Returns proper rotation R = V*diag(1,1,d)*U^T
// with H = U S V^T, d = sign(det(V U^T)) — matches the reference's det-fix on the
// smallest singular direction (eigenvalues sorted descending).
__device__ __forceinline__ void kabsch3(const float H[3][3], float Rm[3][3]) {
    // S = H^T H (symmetric PSD)
    float S[3][3];
#pragma unroll
    for (int i = 0; i < 3; ++i)
#pragma unroll
        for (int j = 0; j < 3; ++j) {
            float a = 0.f;
#pragma unroll
            for (int k = 0; k < 3; ++k) a += H[k][i] * H[k][j];
            S[i][j] = a;
        }
    float V[3][3] = {{1.f,0.f,0.f},{0.f,1.f,0.f},{0.f,0.f,1.f}};
    // cyclic Jacobi sweeps; fully unrolled so p,q are compile-time (register arrays)
#pragma unroll
    for (int sweep = 0; sweep < 6; ++sweep) {
#pragma unroll
        for (int pq = 0; pq < 3; ++pq) {
            const int p = (pq == 2) ? 1 : 0;
            const int q = (pq == 0) ? 1 : 2;
            float apq = S[p][q];
            float c = 1.f, s = 0.f;
            if (fabsf(apq) > 1e-30f) {
                float tau = (S[q][q] - S[p][p]) / (2.f * apq);
                float t   = copysignf(1.f, tau) / (fabsf(tau) + sqrtf(1.f + tau * tau));
                c = rsqrtf(1.f + t * t);
                s = t * c;
            }
#pragma unroll
            for (int k = 0; k < 3; ++k) { float a = S[k][p], b = S[k][q]; S[k][p] = c*a - s*b; S[k][q] = s*a + c*b; }
#pragma unroll
            for (int k = 0; k < 3; ++k) { float a = S[p][k], b = S[q][k]; S[p][k] = c*a - s*b; S[q][k] = s*a + c*b; }
#pragma unroll
            for (int k = 0; k < 3; ++k) { float a = V[k][p], b = V[k][q]; V[k][p] = c*a - s*b; V[k][q] = s*a + c*b; }
        }
    }
    float wv[3] = {S[0][0], S[1][1], S[2][2]};
    // sort eigenvalues descending, permuting V columns (3-element sorting network)
#define CSWAP(a,b)                                                              \
    if (wv[a] < wv[b]) {                                                        \
        float t0 = wv[a]; wv[a] = wv[b]; wv[b] = t0;                            \
        _Pragma("unroll")                                                       \
        for (int k = 0; k < 3; ++k) { float t1 = V[k][a]; V[k][a] = V[k][b]; V[k][b] = t1; } \
    }
    CSWAP(0,1) CSWAP(1,2) CSWAP(0,1)
#undef CSWAP
    // U columns: u_k = H v_k / |H v_k| for k=0,1 ; u_2 = u_0 x u_1 (det(U)=+1)
    float U[3][3];
#pragma unroll
    for (int k = 0; k < 2; ++k) {
        float u0 = H[0][0]*V[0][k] + H[0][1]*V[1][k] + H[0][2]*V[2][k];
        float u1 = H[1][0]*V[0][k] + H[1][1]*V[1][k] + H[1][2]*V[2][k];
        float u2 = H[2][0]*V[0][k] + H[2][1]*V[1][k] + H[2][2]*V[2][k];
        float inv = rsqrtf(fmaxf(u0*u0 + u1*u1 + u2*u2, 1e-24f));
        U[0][k] = u0 * inv; U[1][k] = u1 * inv; U[2][k] = u2 * inv;
    }
    {
        float c0 = U[1][0]*U[2][1] - U[2][0]*U[1][1];
        float c1 = U[2][0]*U[0][1] - U[0][0]*U[2][1];
        float c2 = U[0][0]*U[1][1] - U[1][0]*U[0][1];
        float inv = rsqrtf(fmaxf(c0*c0 + c1*c1 + c2*c2, 1e-24f));
        U[0][2] = c0 * inv; U[1][2] = c1 * inv; U[2][2] = c2 * inv;
    }
    float dV = V[0][0]*(V[1][1]*V[2][2] - V[1][2]*V[2][1])
             - V[0][1]*(V[1][0]*V[2][2] - V[1][2]*V[2][0])
             + V[0][2]*(V[1][0]*V[2][1] - V[1][1]*V[2][0]);
    float d = (dV < 0.f) ? -1.f : 1.f;   // det(U) == +1 by construction
#pragma unroll
    for (int i = 0; i < 3; ++i)
#pragma unroll
        for (int j = 0; j < 3; ++j)
            Rm[i][j] = V[i][0]*U[j][0] + V[i][1]*U[j][1] + d * V[i][2]*U[j][2];
}

// ---------------- small helper kernels ----------------
__global__ void zero_kernel(float* __restrict__ out, float* __restrict__ counts,
                            long nOut, int N) {
    long i = (long)blockIdx.x * blockDim.x + threadIdx.x;
    if (i < nOut) out[i] = 0.f;
    if (i < N)   counts[i] = 0.f;
}

__global__ void counts_kernel(const int* __restrict__ groups, float* __restrict__ counts,
                              int totalG) {
    int i = blockIdx.x * blockDim.x + threadIdx.x;
    if (i < totalG) atomicAdd(&counts[groups[i]], 1.0f);
}

__global__ void finalize_kernel(float* __restrict__ out, const float* __restrict__ counts,
                                long nOut, int N) {
    long i = (long)blockIdx.x * blockDim.x + threadIdx.x;
    if (i < nOut) {
        int node = (int)((i / 3) % N);
        out[i] = out[i] / fmaxf(counts[node], 1.0f);
    }
}

// ---------------- main fused kernel: gather -> WMMA H+sums -> SVD -> project/scatter ----
// One wave handles GPW=5 (b,g) pairs. The 16x16 f32 WMMA tile is packed as:
//   rows/cols m = 3p+i  (p = pair-in-wave, i = xyz component) carry Y / X samples,
//   row 15 of A = all-ones, col 15 of B = all-ones (lanes 15/31 read a 1.0-filled
//   LDS region through the same unconditional dual-address DS loads as data lanes).
// After 16 chained K=4 WMMAs over GS=64 samples:
//   D[3p+i][3p+j] = sum_n Y[p][n][i] * X[p][n][j]   (raw covariance blocks)
//   D[3p+i][15]   = sum_n Y[p][n][i]  = SY[p][i]    (free mean sums from the ones col)
//   D[15][3p+j]   = sum_n X[p][n][j]  = SX[p][j]    (free mean sums from the ones row)
__global__ void __launch_bounds__(TPB)
rigid_kernel(const float* __restrict__ x, const float* __restrict__ tmpl,
             const int* __restrict__ groups, float* __restrict__ out,
             int B, int N, int GN) {
    __shared__ int   sIdx[WAVES][GPW][GS];
    __shared__ float sY  [WAVES][GPW][GS][3];
    __shared__ float sX  [WAVES][GPW][GS][3];
    __shared__ float sD  [WAVES][16][16];
    __shared__ float sRt [WAVES][GPW][12];   // R (9) + t (3)
    __shared__ float sOne[NONES];            // read-only 1.0f region for the ones row/col

    const int  tid        = threadIdx.x;
    const int  w          = tid >> 5;
    const int  lane       = tid & 31;
    const long waveGlobal = (long)blockIdx.x * WAVES + w;
    const long pairBase   = waveGlobal * GPW;
    const long totalPairs = (long)B * GN;
    const float invGS     = 1.0f / (float)GS;

    // ---- Stage 1: gather template + x rows for 5 pairs into LDS; fill ones region ----
    for (int i = tid; i < NONES; i += TPB) sOne[i] = 1.0f;
#pragma unroll
    for (int it = 0; it < (GPW * GS) / 32; ++it) {
        int  s = lane + 32 * it;        // 0..319
        int  p = s >> 6;                // pair-in-wave 0..4
        int  n = s & 63;                // sample 0..63
        long pair = pairBase + p;
        int  id = 0;
        float y0 = 0.f, y1 = 0.f, y2 = 0.f, x0 = 0.f, x1 = 0.f, x2 = 0.f;
        if (pair < totalPairs) {
            int b = (int)(pair / GN);
            int g = (int)(pair - (long)b * GN);
            id = groups[g * GS + n];
            const float* tp = tmpl + (size_t)id * 3;
            y0 = tp[0]; y1 = tp[1]; y2 = tp[2];
            const float* xp = x + ((size_t)b * N + id) * 3;
            x0 = xp[0]; x1 = xp[1]; x2 = xp[2];
        }
        sIdx[w][p][n] = id;
        sY[w][p][n][0] = y0; sY[w][p][n][1] = y1; sY[w][p][n][2] = y2;
        sX[w][p][n][0] = x0; sX[w][p][n][1] = x1; sX[w][p][n][2] = x2;
    }
    __syncthreads();

    // ---- Stage 2: WMMA — covariance blocks + mean sums in one tile ----
    // A f32 16x4 layout: lane = m + 16*(k>=2), vgpr = k&1 ; B mirrored.
    // Base pointer selected once per lane => uniform unconditional DS loads in the loop.
    {
        int  m     = lane & 15;
        int  half  = lane >> 4;
        bool ones  = (m == 15);
        int  mc    = ones ? 0 : m;
        int  p     = mc / 3, c = mc - 3 * (mc / 3);
        const float* yBase = ones ? sOne : &sY[w][p][0][c];   // element n at yBase[3*n]
        const float* xBase = ones ? sOne : &sX[w][p][0][c];
        v8f acc = {0.f,0.f,0.f,0.f,0.f,0.f,0.f,0.f};
#pragma unroll
        for (int step = 0; step < GS / 4; ++step) {
            int n0 = 4 * step + 2 * half;        // lane half selects K = {0,1} vs {2,3}
            v2f av, bv;
            av.x = yBase[3 * n0];
            av.y = yBase[3 * n0 + 3];
            bv.x = xBase[3 * n0];
            bv.y = xBase[3 * n0 + 3];
            acc = __builtin_amdgcn_wmma_f32_16x16x4_f32(
                      false, av, false, bv, (short)0, acc, false, false);
        }
        // D layout: lanes 0-15 -> M=r, N=lane ; lanes 16-31 -> M=r+8, N=lane-16
#pragma unroll
        for (int r = 0; r < 8; ++r) sD[w][r + 8 * half][m] = acc[r];
    }
    __syncthreads();

    // ---- Stage 3: per-pair 3x3 Kabsch (one lane per pair) ----
    if (lane < GPW) {
        long pair = pairBase + lane;
        if (pair < totalPairs) {
            int p = lane;
            float sy[3], sx[3];
#pragma unroll
            for (int i = 0; i < 3; ++i) {
                sy[i] = sD[w][3 * p + i][15];    // SY from the ones column of B
                sx[i] = sD[w][15][3 * p + i];    // SX from the ones row of A
            }
            float H[3][3];
#pragma unroll
            for (int i = 0; i < 3; ++i)
#pragma unroll
                for (int j = 0; j < 3; ++j)
                    H[i][j] = sD[w][3 * p + i][3 * p + j] - sy[i] * sx[j] * invGS;
            float Rm[3][3];
            kabsch3(H, Rm);
#pragma unroll
            for (int i = 0; i < 3; ++i) {
                sRt[w][p][3 * i + 0] = Rm[i][0];
                sRt[w][p][3 * i + 1] = Rm[i][1];
                sRt[w][p][3 * i + 2] = Rm[i][2];
                // t = muX - R muY
                sRt[w][p][9 + i] = sx[i] * invGS
                    - (Rm[i][0] * sy[0] + Rm[i][1] * sy[1] + Rm[i][2] * sy[2]) * invGS;
            }
        }
    }
    __syncthreads();

    // ---- Stage 4: project template points and scatter-add ----
#pragma unroll
    for (int it = 0; it < (GPW * GS) / 32; ++it) {
        int  s = lane + 32 * it;
        int  p = s >> 6;
        int  n = s & 63;
        long pair = pairBase + p;
        if (pair < totalPairs) {
            int b = (int)(pair / GN);
            float y0 = sY[w][p][n][0], y1 = sY[w][p][n][1], y2 = sY[w][p][n][2];
            const float* Rt = sRt[w][p];
            float px = Rt[0]*y0 + Rt[1]*y1 + Rt[2]*y2 + Rt[9];
            float py = Rt[3]*y0 + Rt[4]*y1 + Rt[5]*y2 + Rt[10];
            float pz = Rt[6]*y0 + Rt[7]*y1 + Rt[8]*y2 + Rt[11];
            int id = sIdx[w][p][n];
            float* o = out + ((size_t)b * N + id) * 3;
            atomicAdd(o + 0, px);
            atomicAdd(o + 1, py);
            atomicAdd(o + 2, pz);
        }
    }
}

// ---------------- launcher ----------------
extern "C" void kernel_launch(void* const* d_in, const int* in_sizes, int n_in,
                              void* d_out, int out_size, void* d_ws, size_t ws_size,
                              hipStream_t stream) {
    const float* x      = (const float*)d_in[0];   // (B, N, 3) f32
    const float* tmpl   = (const float*)d_in[1];   // (N, 3) f32
    const int*   groups = (const int*)d_in[2];     // (GN, GS) i32
    float*       out    = (float*)d_out;           // (B, N, 3) f32

    const int N      = in_sizes[1] / 3;
    const int B      = in_sizes[0] / (3 * N);
    const int totalG = in_sizes[2];
    const int GN     = totalG / GS;

    float* counts = (float*)d_ws;                  // N floats of scratch

    const long nOut = (long)B * N * 3;
    const long zMax = (nOut > (long)N) ? nOut : (long)N;

    zero_kernel<<<(int)((zMax + 255) / 256), 256, 0, stream>>>(out, counts, nOut, N);
    counts_kernel<<<(totalG + 255) / 256, 256, 0, stream>>>(groups, counts, totalG);

    const long pairs  = (long)B * GN;
    const long waves  = (pairs + GPW - 1) / GPW;
    const int  blocks = (int)((waves + WAVES - 1) / WAVES);
    rigid_kernel<<<blocks, TPB, 0, stream>>>(x, tmpl, groups, out, B, N, GN);

    finalize_kernel<<<(int)((nOut + 255) / 256), 256, 0, stream>>>(out, counts, nOut, N);
}